// BarycentricCoordinates_89610197664487
// MI455X (gfx1250) — compile-verified
//
#include <hip/hip_runtime.h>
#include <math.h>

typedef __attribute__((ext_vector_type(2))) float v2f;
typedef __attribute__((ext_vector_type(8))) float v8f;

#define WAVES_PER_BLOCK 8

__global__ __launch_bounds__(256) void bary_wmma_kernel(
    const float* __restrict__ tmpl,    // (5,8,2) = 80 floats
    const float* __restrict__ proj,    // (V,16,2)
    float* __restrict__ w_out,         // (V,40,3) f32
    int*   __restrict__ idx_out,       // (V,40,3) i32
    int V)
{
  constexpr int N  = 16;
  constexpr int K  = 15;
  constexpr int RA = 40;

  __shared__ float s_tmpl[RA * 2];
  __shared__ float s_proj[N * 2];
  __shared__ float s_G[WAVES_PER_BLOCK][16 * 16];
  __shared__ int   s_hier[WAVES_PER_BLOCK][16];

  const int v    = blockIdx.x;
  const int tid  = threadIdx.x;
  const int wave = tid >> 5;
  const int lane = tid & 31;

  if (tid < RA * 2) s_tmpl[tid] = tmpl[tid];
  if (tid < N * 2)  s_proj[tid] = proj[(size_t)v * (N * 2) + tid];
  __syncthreads();

  // RA=40, 8 waves -> exactly 5 uniform iterations per wave (barrier-safe)
  for (int cell = wave; cell < RA; cell += WAVES_PER_BLOCK) {
    const float tx = s_tmpl[cell * 2 + 0];
    const float ty = s_tmpl[cell * 2 + 1];

    // ---- Phase 1: distances + stable rank sort of 16 neighbors (lanes 0..15)
    float d = INFINITY;
    if (lane < N) {
      const float px = s_proj[lane * 2 + 0];
      const float py = s_proj[lane * 2 + 1];
      const float dx = tx - px, dy = ty - py;
      d = sqrtf(dx * dx + dy * dy);
    }
    int rank = 0;
    #pragma unroll
    for (int j = 0; j < N; ++j) {
      const float dj = __shfl(d, j);
      rank += ((dj < d) || (dj == d && j < lane)) ? 1 : 0;
    }
    if (lane < N) s_hier[wave][rank] = lane;
    __syncthreads();

    // ---- Phase 2: rows = [v0_0..v0_14, v2]; Gram via V_WMMA_F32_16X16X4_F32
    const int   h0 = s_hier[wave][0];
    const float cx = s_proj[h0 * 2 + 0];
    const float cy = s_proj[h0 * 2 + 1];

    float ax = 0.f, ay = 0.f;
    if (lane < K) {                       // rows 0..14: v0_k
      const int hm = s_hier[wave][lane + 1];
      ax = s_proj[hm * 2 + 0] - cx;
      ay = s_proj[hm * 2 + 1] - cy;
    } else if (lane == K) {               // row 15: v2 = t - closest
      ax = tx - cx;
      ay = ty - cy;
    }                                     // lanes 16..31 carry K=2,3 planes -> 0

    v2f aOp;
    aOp[0] = ax;
    aOp[1] = ay;

    v8f cAcc = {};
    v8f g = __builtin_amdgcn_wmma_f32_16x16x4_f32(
        /*neg_a=*/false, aOp, /*neg_b=*/false, aOp,
        /*c_mod=*/(short)0, cAcc, /*reuse_a=*/false, /*reuse_b=*/false);

    // D layout: lane holds column n = lane&15; VGPR k -> row (lane>=16 ? 8 : 0)+k
    {
      const int n     = lane & 15;
      const int mbase = (lane >> 4) * 8;
      #pragma unroll
      for (int k = 0; k < 8; ++k)
        s_G[wave][(mbase + k) * 16 + n] = g[k];
    }
    __syncthreads();

    // ---- Phase 3: scan 225 (i,m) pairs; argmin of metric (first occurrence)
    float bm = INFINITY;
    int   bi = K * K;                     // sentinel
    float bw0 = 0.f, bw1 = 0.f, bw2 = 0.f;
    const float* Gw = s_G[wave];
    for (int p = lane; p < K * K; p += 32) {
      const int i = p / K;
      const int m = p - i * K;
      const float d00i = Gw[i * 16 + i];
      const float d00m = Gw[m * 16 + m];
      const float d01  = Gw[i * 16 + m];
      const float d02i = Gw[i * 16 + 15];
      const float d02m = Gw[m * 16 + 15];
      float denom = d00i * d00m - d01 * d01;
      denom = (denom == 0.f) ? 1e-10f : denom;
      const float p2v = (d02i * d00m - d01 * d02m) / denom;
      const float p1v = (d00i * d02m - d01 * d02i) / denom;
      const float p0v = 1.f - p2v - p1v;
      const float w0 = (p0v <= 0.f) ? INFINITY : p0v;
      const float w1 = (p2v <= 0.f) ? INFINITY : p2v;
      const float w2 = (p1v <= 0.f) ? INFINITY : p1v;
      const float metric = fmaxf(fmaxf(w0 * w0, w1 * w1), w2 * w2);
      if ((metric < bm) || (metric == bm && p < bi)) {
        bm = metric; bi = p; bw0 = w0; bw1 = w1; bw2 = w2;
      }
    }
    // wave32 lexicographic-min reduction over (metric, flat_idx)
    #pragma unroll
    for (int off = 16; off > 0; off >>= 1) {
      const float om = __shfl_xor(bm,  off);
      const int   oi = __shfl_xor(bi,  off);
      const float o0 = __shfl_xor(bw0, off);
      const float o1 = __shfl_xor(bw1, off);
      const float o2 = __shfl_xor(bw2, off);
      if ((om < bm) || (om == bm && oi < bi)) {
        bm = om; bi = oi; bw0 = o0; bw1 = o1; bw2 = o2;
      }
    }

    if (lane == 0) {
      if (bi >= K * K) bi = 0;            // degenerate all-NaN guard
      const int row = bi / K;
      const int col = bi - row * K;
      const bool anyInf = (bw0 == INFINITY) || (bw1 == INFINITY) || (bw2 == INFINITY);
      const float o0 = anyInf ? 0.f : bw0;
      const float o1 = anyInf ? 0.f : bw1;
      const float o2 = anyInf ? 0.f : bw2;
      const size_t base = ((size_t)v * RA + cell) * 3;
      w_out[base + 0] = o0;
      w_out[base + 1] = o1;
      w_out[base + 2] = o2;
      idx_out[base + 0] = h0;
      idx_out[base + 1] = s_hier[wave][row + 1];
      idx_out[base + 2] = s_hier[wave][col + 1];
    }
  }
}

extern "C" void kernel_launch(void* const* d_in, const int* in_sizes, int n_in,
                              void* d_out, int out_size, void* d_ws, size_t ws_size,
                              hipStream_t stream) {
  const float* tmpl = (const float*)d_in[0];   // (5,8,2)
  const float* proj = (const float*)d_in[1];   // (V,16,2)
  const int V = in_sizes[1] / 32;              // 16 neighbors * 2 comps
  float* w_out  = (float*)d_out;                         // V*40*3 floats
  int*   idx_out = (int*)d_out + (size_t)V * 40 * 3;     // then V*40*3 int32
  bary_wmma_kernel<<<V, 256, 0, stream>>>(tmpl, proj, w_out, idx_out, V);
}